// SyntheticTripletLossSmoothL1Sentence_9861244912211
// MI455X (gfx1250) — compile-verified
//
#include <hip/hip_runtime.h>
#include <cstdint>

// Problem constants (match reference)
#define BB     64
#define LL     512
#define DD     1024
#define SPLIT  16
#define ROWS   (LL / SPLIT)     // 32 rows per block
#define TPB    256              // 8 wave32 waves; DD / TPB = 4 floats per lane
#define PIPE   4                // async pipeline depth (rows in flight)
#define MARGIN 0.5f
#define BETA   0.5f
#define EPSN   1e-12f

// ---------------------------------------------------------------------------
// CDNA5 async global->LDS copy (gfx1250). One 128b element per lane.
// VDST = LDS byte address (low 32 bits of the flat shared pointer),
// VADDR = 64-bit global address, no SADDR ("off"). Tracked by ASYNCcnt.
// ---------------------------------------------------------------------------
__device__ __forceinline__ void async_copy16(const float* gsrc, const float* ldst) {
  unsigned lds = (unsigned)(uintptr_t)ldst;              // flat->LDS offset (addr[31:0])
  unsigned long long ga = (unsigned long long)(uintptr_t)gsrc;
  asm volatile("global_load_async_to_lds_b128 %0, %1, off"
               :: "v"(lds), "v"(ga)
               : "memory");
}
#define ASYNC_WAIT(n) asm volatile("s_wait_asynccnt " #n ::: "memory")

__device__ __forceinline__ float sl1(float a, float b) {
  float d = a - b;
  float ad = fabsf(d);
  return (ad < 1.0f) ? (0.5f * d * d) : (ad - 0.5f);
}

// Block-wide sum of three values, broadcast to all lanes. 8 waves, wave32.
__device__ __forceinline__ void block_reduce3(float& a, float& b, float& c,
                                              volatile float* red) {
#pragma unroll
  for (int m = 16; m > 0; m >>= 1) {
    a += __shfl_xor(a, m, 32);
    b += __shfl_xor(b, m, 32);
    c += __shfl_xor(c, m, 32);
  }
  const int lane = threadIdx.x & 31;
  const int wid  = threadIdx.x >> 5;
  if (lane == 0) { red[wid] = a; red[8 + wid] = b; red[16 + wid] = c; }
  __syncthreads();
  float ra = 0.f, rb = 0.f, rc = 0.f;
#pragma unroll
  for (int i = 0; i < 8; ++i) { ra += red[i]; rb += red[8 + i]; rc += red[16 + i]; }
  __syncthreads();
  a = ra; b = rb; c = rc;
}

__device__ __forceinline__ void block_reduce2(float& a, float& b, volatile float* red) {
#pragma unroll
  for (int m = 16; m > 0; m >>= 1) {
    a += __shfl_xor(a, m, 32);
    b += __shfl_xor(b, m, 32);
  }
  const int lane = threadIdx.x & 31;
  const int wid  = threadIdx.x >> 5;
  if (lane == 0) { red[wid] = a; red[8 + wid] = b; }
  __syncthreads();
  float ra = 0.f, rb = 0.f;
#pragma unroll
  for (int i = 0; i < 8; ++i) { ra += red[i]; rb += red[8 + i]; }
  __syncthreads();
  a = ra; b = rb;
}

__device__ __forceinline__ float block_reduce1(float a, volatile float* red) {
#pragma unroll
  for (int m = 16; m > 0; m >>= 1) a += __shfl_xor(a, m, 32);
  const int lane = threadIdx.x & 31;
  const int wid  = threadIdx.x >> 5;
  __syncthreads();
  if (lane == 0) red[wid] = a;
  __syncthreads();
  float r = 0.f;
#pragma unroll
  for (int i = 0; i < 8; ++i) r += red[i];
  return r;
}

// ---------------------------------------------------------------------------
// Kernel 1: per-row triplet terms + masked partial sums for one 32-row chunk.
// 4-deep async global->LDS pipeline; deterministic per-(b,split) partials.
// ---------------------------------------------------------------------------
__global__ void __launch_bounds__(TPB) triplet_rows_kernel(
    const float* __restrict__ preds, const float* __restrict__ targets,
    const int* __restrict__ lens,
    float* __restrict__ pm, float* __restrict__ tm, float* __restrict__ nm,
    float* __restrict__ wordp)
{
  __shared__ float sp[PIPE][DD];   // 4 x 4 KB ring buffer (preds rows)
  __shared__ float st[PIPE][DD];   // 4 x 4 KB ring buffer (target rows)
  __shared__ float red[24];

  const int b   = blockIdx.x;
  const int s   = blockIdx.y;
  const int tid = threadIdx.x;
  const int len = lens[b];
  const int l0  = s * ROWS;
  int l1 = l0 + ROWS;
  if (len < l1) l1 = len;      // mask: skip invalid rows entirely (uniform per block)

  float ap0 = 0.f, ap1 = 0.f, ap2 = 0.f, ap3 = 0.f;   // masked sum of preds
  float at0 = 0.f, at1 = 0.f, at2 = 0.f, at3 = 0.f;   // masked sum of targets
  float an0 = 0.f, an1 = 0.f, an2 = 0.f, an3 = 0.f;   // masked sum of neg
  float wsum = 0.f;                                   // masked word-loss sum

  if (l0 < l1) {
    const size_t rowbase = (size_t)b * LL * DD + (size_t)tid * 4;
    const unsigned loff  = (unsigned)tid * 4u;

    // Prologue: stage up to PIPE rows asynchronously.
#pragma unroll
    for (int i = 0; i < PIPE; ++i) {
      if (l0 + i < l1) {
        async_copy16(preds   + rowbase + (size_t)(l0 + i) * DD, &sp[i][loff]);
        async_copy16(targets + rowbase + (size_t)(l0 + i) * DD, &st[i][loff]);
      }
    }

    for (int l = l0; l < l1; ++l) {
      const int buf = (l - l0) & (PIPE - 1);
      // Outstanding async instructions before this wait: 2*min(PIPE, rem).
      // Loads retire in order, so waiting to 2*min(PIPE,rem)-2 guarantees the
      // current row's two loads have landed in LDS.
      const int rem = l1 - l;
      if (rem >= 4)      { ASYNC_WAIT(6); }
      else if (rem == 3) { ASYNC_WAIT(4); }
      else if (rem == 2) { ASYNC_WAIT(2); }
      else               { ASYNC_WAIT(0); }

      const float4 p = *(const float4*)&sp[buf][loff];
      const float4 t = *(const float4*)&st[buf][loff];

      float dpt = p.x * t.x + p.y * t.y + p.z * t.z + p.w * t.w;
      float ppv = p.x * p.x + p.y * p.y + p.z * p.z + p.w * p.w;
      float ttv = t.x * t.x + t.y * t.y + t.z * t.z + t.w * t.w;
      block_reduce3(dpt, ppv, ttv, red);   // dot, |p|^2, |t|^2 (one barrier pair)

      // ||p - dot*t||^2 = pp - 2*dot^2 + dot^2*tt
      float norm2 = fmaxf(ppv + dpt * dpt * (ttv - 2.0f), 0.0f);
      float inv   = 1.0f / fmaxf(sqrtf(norm2), EPSN);

      float nx, lt, ln;
      nx = (p.x - dpt * t.x) * inv; lt = sl1(p.x, t.x); ln = sl1(p.x, nx);
      wsum += fmaxf(MARGIN + lt - ln, 0.f); ap0 += p.x; at0 += t.x; an0 += nx;
      nx = (p.y - dpt * t.y) * inv; lt = sl1(p.y, t.y); ln = sl1(p.y, nx);
      wsum += fmaxf(MARGIN + lt - ln, 0.f); ap1 += p.y; at1 += t.y; an1 += nx;
      nx = (p.z - dpt * t.z) * inv; lt = sl1(p.z, t.z); ln = sl1(p.z, nx);
      wsum += fmaxf(MARGIN + lt - ln, 0.f); ap2 += p.z; at2 += t.z; an2 += nx;
      nx = (p.w - dpt * t.w) * inv; lt = sl1(p.w, t.w); ln = sl1(p.w, nx);
      wsum += fmaxf(MARGIN + lt - ln, 0.f); ap3 += p.w; at3 += t.w; an3 += nx;

      // Refill the buffer we just fully consumed with row l+PIPE.
      if (l + PIPE < l1) {
        async_copy16(preds   + rowbase + (size_t)(l + PIPE) * DD, &sp[buf][loff]);
        async_copy16(targets + rowbase + (size_t)(l + PIPE) * DD, &st[buf][loff]);
      }
    }
  }

  // Deterministic partials: one 128b store per array per thread.
  const size_t obase = (size_t)(b * SPLIT + s) * DD + (size_t)tid * 4;
  float4 o;
  o.x = ap0; o.y = ap1; o.z = ap2; o.w = ap3; *(float4*)&pm[obase] = o;
  o.x = at0; o.y = at1; o.z = at2; o.w = at3; *(float4*)&tm[obase] = o;
  o.x = an0; o.y = an1; o.z = an2; o.w = an3; *(float4*)&nm[obase] = o;

  const float wtot = block_reduce1(wsum, red);
  if (tid == 0) wordp[b * SPLIT + s] = wtot;
}

// ---------------------------------------------------------------------------
// Kernel 2: per-sentence combine. Sum the 16 split-partials (deterministic),
// form mean embeddings, sentence triplet, and contrib[b].
// ---------------------------------------------------------------------------
__global__ void __launch_bounds__(TPB) sent_kernel(
    const float* __restrict__ pm, const float* __restrict__ tm,
    const float* __restrict__ nm, const float* __restrict__ wordp,
    const int* __restrict__ lens, float* __restrict__ contrib)
{
  __shared__ float red[16];
  const int b   = blockIdx.x;
  const int tid = threadIdx.x;
  const float len    = (float)lens[b];
  const float invlen = 1.0f / len;

  float sp0 = 0.f, sp1 = 0.f, sp2 = 0.f, sp3 = 0.f;
  float st0 = 0.f, st1 = 0.f, st2 = 0.f, st3 = 0.f;
  float sn0 = 0.f, sn1 = 0.f, sn2 = 0.f, sn3 = 0.f;
#pragma unroll
  for (int s = 0; s < SPLIT; ++s) {
    const size_t base = (size_t)(b * SPLIT + s) * DD + (size_t)tid * 4;
    float4 a = *(const float4*)&pm[base];
    float4 c = *(const float4*)&tm[base];
    float4 e = *(const float4*)&nm[base];
    sp0 += a.x; sp1 += a.y; sp2 += a.z; sp3 += a.w;
    st0 += c.x; st1 += c.y; st2 += c.z; st3 += c.w;
    sn0 += e.x; sn1 += e.y; sn2 += e.z; sn3 += e.w;
  }

  float ssum = 0.f;
  {
    float p, t, n;
    p = sp0 * invlen; t = st0 * invlen; n = sn0 * invlen;
    ssum += fmaxf(MARGIN + sl1(p, t) - sl1(p, n), 0.f);
    p = sp1 * invlen; t = st1 * invlen; n = sn1 * invlen;
    ssum += fmaxf(MARGIN + sl1(p, t) - sl1(p, n), 0.f);
    p = sp2 * invlen; t = st2 * invlen; n = sn2 * invlen;
    ssum += fmaxf(MARGIN + sl1(p, t) - sl1(p, n), 0.f);
    p = sp3 * invlen; t = st3 * invlen; n = sn3 * invlen;
    ssum += fmaxf(MARGIN + sl1(p, t) - sl1(p, n), 0.f);
  }

  float wv = (tid < SPLIT) ? wordp[b * SPLIT + tid] : 0.f;
  block_reduce2(ssum, wv, red);

  if (tid == 0) {
    const float word_per_sent = wv / (len * (float)DD);
    const float sent_per_sent = ssum / (float)DD;
    contrib[b] = (1.0f - BETA) * word_per_sent + BETA * sent_per_sent;
  }
}

// ---------------------------------------------------------------------------
// Kernel 3: mean over B sentences -> scalar output.
// ---------------------------------------------------------------------------
__global__ void __launch_bounds__(64) final_kernel(const float* __restrict__ contrib,
                                                   float* __restrict__ out)
{
  __shared__ float red[2];
  const int tid = threadIdx.x;   // exactly 64 threads = 2 waves
  float v = contrib[tid];
#pragma unroll
  for (int m = 16; m > 0; m >>= 1) v += __shfl_xor(v, m, 32);
  if ((tid & 31) == 0) red[tid >> 5] = v;
  __syncthreads();
  if (tid == 0) out[0] = (red[0] + red[1]) * (1.0f / (float)BB);
}

// ---------------------------------------------------------------------------
extern "C" void kernel_launch(void* const* d_in, const int* in_sizes, int n_in,
                              void* d_out, int out_size, void* d_ws, size_t ws_size,
                              hipStream_t stream) {
  (void)in_sizes; (void)n_in; (void)out_size; (void)ws_size;
  const float* preds   = (const float*)d_in[0];
  const float* targets = (const float*)d_in[1];
  const int*   lens    = (const int*)d_in[2];
  float* out = (float*)d_out;

  // Workspace layout (floats): 3 partial-sum tensors + word partials + contrib.
  float* ws      = (float*)d_ws;
  const size_t P = (size_t)BB * SPLIT * DD;   // 1,048,576 floats each
  float* pm      = ws;
  float* tm      = pm + P;
  float* nm      = tm + P;
  float* wordp   = nm + P;              // BB*SPLIT floats
  float* contrib = wordp + BB * SPLIT;  // BB floats
  // Total: ~12.6 MB, all unconditionally written every launch.

  triplet_rows_kernel<<<dim3(BB, SPLIT), TPB, 0, stream>>>(
      preds, targets, lens, pm, tm, nm, wordp);
  sent_kernel<<<BB, TPB, 0, stream>>>(pm, tm, nm, wordp, lens, contrib);
  final_kernel<<<1, 64, 0, stream>>>(contrib, out);
}